// HopfieldGridRnn_38706245272466
// MI455X (gfx1250) — compile-verified
//
#include <hip/hip_runtime.h>
#include <math.h>

// ---------------------------------------------------------------------------
// Sizes (fixed by the reference): V=50257, E=512, H=512, O=50257, L=4, C=8,
// NH=8, B=64.  All data f32.  All matmuls via V_WMMA_F32_16X16X4_F32 (full
// f32 precision; workload is HBM-bandwidth-bound so f32 WMMA is optimal).
// ---------------------------------------------------------------------------

typedef float v2f __attribute__((ext_vector_type(2)));
typedef float v8f __attribute__((ext_vector_type(8)));

__device__ __forceinline__ v8f wmma4(v2f a, v2f b, v8f c) {
  // D = A(16x4, f32) x B(4x16, f32) + C(16x16, f32)
  return __builtin_amdgcn_wmma_f32_16x16x4_f32(
      /*neg_a=*/false, a, /*neg_b=*/false, b,
      /*c_mod=*/(short)0, c, /*reuse_a=*/false, /*reuse_b=*/false);
}

// Strip GEMM helpers: one wave computes a 64(M) x 16(N) strip, K=512.
// A rows are row-major with ld=512; "a1" points at row (sub) col (2*half).
// W is (N,512) row-major; "w1" points at row (n0+sub) col (2*half); B = W^T.
// B fragment is loaded once and reused by 4 stacked M-tiles.
__device__ __forceinline__ void strip4_single(const float* __restrict__ a1,
                                              const float* __restrict__ w1,
                                              v8f acc[4]) {
  for (int kk = 0; kk < 512; kk += 4) {
    v2f bv; bv.x = w1[kk]; bv.y = w1[kk + 1];
#pragma unroll
    for (int mt = 0; mt < 4; ++mt) {
      v2f av; av.x = a1[mt * 8192 + kk]; av.y = a1[mt * 8192 + kk + 1];
      acc[mt] = wmma4(av, bv, acc[mt]);
    }
  }
}

__device__ __forceinline__ void strip4_dual(const float* __restrict__ a1,
                                            const float* __restrict__ a2,
                                            const float* __restrict__ w1,
                                            const float* __restrict__ w2,
                                            v8f acc[4]) {
  for (int kk = 0; kk < 512; kk += 4) {
    v2f b1; b1.x = w1[kk]; b1.y = w1[kk + 1];
    v2f b2; b2.x = w2[kk]; b2.y = w2[kk + 1];
#pragma unroll
    for (int mt = 0; mt < 4; ++mt) {
      v2f av; av.x = a1[mt * 8192 + kk]; av.y = a1[mt * 8192 + kk + 1];
      acc[mt] = wmma4(av, b1, acc[mt]);
      v2f aw; aw.x = a2[mt * 8192 + kk]; aw.y = a2[mt * 8192 + kk + 1];
      acc[mt] = wmma4(aw, b2, acc[mt]);
    }
  }
}

// ---------------------------------------------------------------------------
// Embedding gather: x[b][e] = emb[tok[b]][e]   (64 x 512)
// ---------------------------------------------------------------------------
__global__ void k_embed(const int* __restrict__ tok,
                        const float* __restrict__ emb,
                        float* __restrict__ x) {
  int idx = blockIdx.x * 256 + threadIdx.x;  // 32768
  int b = idx >> 9, e = idx & 511;
  x[idx] = emb[(size_t)tok[b] * 512 + e];
}

// ---------------------------------------------------------------------------
// Layer-0 gates.  c==0: g = x@Wih0^T + h0[0,0]@Whh0^T + bih0 + bhh0
//                 c>0 : g = h0[0,c]@Whh0r[c-1]^T + bih0r[c-1] + bhh0r[c-1]
// gates laid out as (C*B, 2048) rows.
// grid = (16, 8), block = 256 (8 waves, one 64Mx16N strip each)
// ---------------------------------------------------------------------------
__global__ void k_gates_l0(const float* __restrict__ x,
                           const float* __restrict__ h0l0,
                           const float* __restrict__ Wih0,
                           const float* __restrict__ Whh0,
                           const float* __restrict__ bih0,
                           const float* __restrict__ bhh0,
                           const float* __restrict__ Whh0r,
                           const float* __restrict__ bih0r,
                           const float* __restrict__ bhh0r,
                           float* __restrict__ gates) {
  int wv = threadIdx.x >> 5, lane = threadIdx.x & 31;
  int nt = blockIdx.x * 8 + wv;  // 0..127
  int c = blockIdx.y;            // 0..7
  int sub = lane & 15, half = lane >> 4;
  int n = nt * 16 + sub;         // 0..2047
  v8f zero = {};
  v8f acc[4] = {zero, zero, zero, zero};
  float bias;
  if (c == 0) {
    strip4_dual(x + sub * 512 + 2 * half,
                h0l0 + sub * 512 + 2 * half,
                Wih0 + (size_t)n * 512 + 2 * half,
                Whh0 + (size_t)n * 512 + 2 * half, acc);
    bias = bih0[n] + bhh0[n];
  } else {
    strip4_single(h0l0 + (size_t)c * 32768 + sub * 512 + 2 * half,
                  Whh0r + ((size_t)(c - 1) * 2048 + n) * 512 + 2 * half, acc);
    bias = bih0r[(c - 1) * 2048 + n] + bhh0r[(c - 1) * 2048 + n];
  }
#pragma unroll
  for (int mt = 0; mt < 4; ++mt)
#pragma unroll
    for (int r = 0; r < 8; ++r)
      gates[(size_t)(c * 64 + mt * 16 + r + 8 * half) * 2048 + n] =
          acc[mt][r] + bias;
}

// ---------------------------------------------------------------------------
// Layer l>0 gates: g[c] = hl[c]@Wih[l-1,c]^T + h0[l,c]@Whh[l-1,c]^T + biases
// ---------------------------------------------------------------------------
__global__ void k_gates_dual(const float* __restrict__ hlw,
                             const float* __restrict__ h0l,
                             const float* __restrict__ Wihl,
                             const float* __restrict__ Whhl,
                             const float* __restrict__ bihl,
                             const float* __restrict__ bhhl,
                             float* __restrict__ gates) {
  int wv = threadIdx.x >> 5, lane = threadIdx.x & 31;
  int nt = blockIdx.x * 8 + wv;
  int c = blockIdx.y;
  int sub = lane & 15, half = lane >> 4;
  int n = nt * 16 + sub;
  v8f zero = {};
  v8f acc[4] = {zero, zero, zero, zero};
  strip4_dual(hlw + (size_t)c * 32768 + sub * 512 + 2 * half,
              h0l + (size_t)c * 32768 + sub * 512 + 2 * half,
              Wihl + ((size_t)c * 2048 + n) * 512 + 2 * half,
              Whhl + ((size_t)c * 2048 + n) * 512 + 2 * half, acc);
  float bias = bihl[c * 2048 + n] + bhhl[c * 2048 + n];
#pragma unroll
  for (int mt = 0; mt < 4; ++mt)
#pragma unroll
    for (int r = 0; r < 8; ++r)
      gates[(size_t)(c * 64 + mt * 16 + r + 8 * half) * 2048 + n] =
          acc[mt][r] + bias;
}

// ---------------------------------------------------------------------------
// LSTM pointwise: (i,f,g,o) gate order, c2 = sig(f)*c0 + sig(i)*tanh(g),
// h = sig(o)*tanh(c2).  262144 elements.
// ---------------------------------------------------------------------------
__global__ void k_lstm(const float* __restrict__ gates,
                       const float* __restrict__ c0l,
                       float* __restrict__ hl,
                       float* __restrict__ cnl) {
  int idx = blockIdx.x * 256 + threadIdx.x;
  int cb = idx >> 9, j = idx & 511;
  const float* g = gates + (size_t)cb * 2048;
  float i_ = 1.f / (1.f + expf(-g[j]));
  float f_ = 1.f / (1.f + expf(-g[512 + j]));
  float gg = tanhf(g[1024 + j]);
  float o_ = 1.f / (1.f + expf(-g[1536 + j]));
  float c2 = f_ * c0l[idx] + i_ * gg;
  cnl[idx] = c2;
  hl[idx] = o_ * tanhf(c2);
}

// ---------------------------------------------------------------------------
// 512x512x512 strip GEMM: out = A @ W^T (+ bias).
// One wave per 64Mx16N strip: 8 m-strips x 32 n-tiles = 256 waves,
// grid.x = 32, block = 256.  B fragment reused across 4 M-tiles.
// ---------------------------------------------------------------------------
__global__ void k_proj512(const float* __restrict__ A,
                          const float* __restrict__ W,
                          const float* __restrict__ bias,
                          float* __restrict__ out) {
  int wv = threadIdx.x >> 5, lane = threadIdx.x & 31;
  int wid = blockIdx.x * 8 + wv;   // 0..255
  int nt = wid & 31, ms = wid >> 5;
  int sub = lane & 15, half = lane >> 4;
  int n = nt * 16 + sub;
  v8f zero = {};
  v8f acc[4] = {zero, zero, zero, zero};
  strip4_single(A + (size_t)(ms * 64 + sub) * 512 + 2 * half,
                W + (size_t)n * 512 + 2 * half, acc);
  float bb = bias ? bias[n] : 0.f;
#pragma unroll
  for (int mt = 0; mt < 4; ++mt)
#pragma unroll
    for (int r = 0; r < 8; ++r)
      out[(size_t)(ms * 64 + mt * 16 + r + 8 * half) * 512 + n] =
          acc[mt][r] + bb;
}

// q/k/v projections fused: blockIdx.y selects which (W, out) pair.
__global__ void k_qkv(const float* __restrict__ A,
                      const float* __restrict__ Wq,
                      const float* __restrict__ Wk,
                      const float* __restrict__ Wv,
                      float* __restrict__ q, float* __restrict__ k,
                      float* __restrict__ v) {
  const float* W = blockIdx.y == 0 ? Wq : (blockIdx.y == 1 ? Wk : Wv);
  float* out = blockIdx.y == 0 ? q : (blockIdx.y == 1 ? k : v);
  int wv = threadIdx.x >> 5, lane = threadIdx.x & 31;
  int wid = blockIdx.x * 8 + wv;   // 0..255
  int nt = wid & 31, ms = wid >> 5;
  int sub = lane & 15, half = lane >> 4;
  int n = nt * 16 + sub;
  v8f zero = {};
  v8f acc[4] = {zero, zero, zero, zero};
  strip4_single(A + (size_t)(ms * 64 + sub) * 512 + 2 * half,
                W + (size_t)n * 512 + 2 * half, acc);
#pragma unroll
  for (int mt = 0; mt < 4; ++mt)
#pragma unroll
    for (int r = 0; r < 8; ++r)
      out[(size_t)(ms * 64 + mt * 16 + r + 8 * half) * 512 + n] = acc[mt][r];
}

// ---------------------------------------------------------------------------
// Hopfield attention: per (head h, batch b): scores (8x8) = beta*q.k^T,
// softmax over e, out = attn @ v.  q/k/v/ao are (C,B,512) with head slice h*64.
// grid = NH*B = 512 blocks, block = 64.
// ---------------------------------------------------------------------------
__global__ void k_attn(const float* __restrict__ q,
                       const float* __restrict__ k,
                       const float* __restrict__ v,
                       const float* __restrict__ lb,
                       float* __restrict__ ao) {
  int h = blockIdx.x >> 6, b = blockIdx.x & 63;
  int t = threadIdx.x;  // 0..63
  __shared__ float sS[8][8];
  int c = t >> 3, e = t & 7;
  const float* qp = q + (size_t)(c * 64 + b) * 512 + h * 64;
  const float* kp = k + (size_t)(e * 64 + b) * 512 + h * 64;
  float acc = 0.f;
  for (int d = 0; d < 64; ++d) acc += qp[d] * kp[d];
  float beta = expf(lb[h]);
  sS[c][e] = beta * acc;
  __syncthreads();
  float m = sS[c][0];
  for (int j = 1; j < 8; ++j) m = fmaxf(m, sS[c][j]);
  float den = 0.f;
  for (int j = 0; j < 8; ++j) den += expf(sS[c][j] - m);
  float w = expf(sS[c][e] - m) / den;
  __syncthreads();
  sS[c][e] = w;
  __syncthreads();
  for (int cc = 0; cc < 8; ++cc) {
    float o = 0.f;
    for (int ee = 0; ee < 8; ++ee)
      o += sS[cc][ee] * v[(size_t)(ee * 64 + b) * 512 + h * 64 + t];
    ao[(size_t)(cc * 64 + b) * 512 + h * 64 + t] = o;
  }
}

// ---------------------------------------------------------------------------
// LayerNorm over last dim (512) of (C*B, 512); grid = 512 rows, block = 256.
// ---------------------------------------------------------------------------
__global__ void k_ln(const float* __restrict__ op,
                     const float* __restrict__ g,
                     const float* __restrict__ bpar,
                     float* __restrict__ msg) {
  int row = blockIdx.x;
  int t = threadIdx.x;
  const float* xr = op + (size_t)row * 512;
  __shared__ float sb[256];
  sb[t] = xr[t] + xr[t + 256];
  __syncthreads();
  for (int o = 128; o > 0; o >>= 1) {
    if (t < o) sb[t] += sb[t + o];
    __syncthreads();
  }
  float mean = sb[0] * (1.f / 512.f);
  __syncthreads();
  float d0 = xr[t] - mean, d1 = xr[t + 256] - mean;
  sb[t] = d0 * d0 + d1 * d1;
  __syncthreads();
  for (int o = 128; o > 0; o >>= 1) {
    if (t < o) sb[t] += sb[t + o];
    __syncthreads();
  }
  float rstd = rsqrtf(sb[0] * (1.f / 512.f) + 1e-5f);
  msg[(size_t)row * 512 + t] = d0 * rstd * g[t] + bpar[t];
  msg[(size_t)row * 512 + t + 256] = d1 * rstd * g[t + 256] + bpar[t + 256];
}

// ---------------------------------------------------------------------------
// Gated mix: gate = sigmoid([h,msg] . Wg + bg) (scalar per row),
// h = (1-gate)*h + gate*msg; also writes h_n output region.
// grid = 512 rows, block = 256.
// ---------------------------------------------------------------------------
__global__ void k_mix(float* __restrict__ hl,
                      const float* __restrict__ msg,
                      const float* __restrict__ Wg,
                      const float* __restrict__ bg,
                      float* __restrict__ hnl) {
  int row = blockIdx.x;
  int t = threadIdx.x;
  float* h = hl + (size_t)row * 512;
  const float* m = msg + (size_t)row * 512;
  __shared__ float sb[256];
  sb[t] = h[t] * Wg[t] + h[t + 256] * Wg[t + 256] + m[t] * Wg[512 + t] +
          m[t + 256] * Wg[768 + t];
  __syncthreads();
  for (int o = 128; o > 0; o >>= 1) {
    if (t < o) sb[t] += sb[t + o];
    __syncthreads();
  }
  float gate = 1.f / (1.f + expf(-(sb[0] + bg[0])));
  float a0 = (1.f - gate) * h[t] + gate * m[t];
  float a1 = (1.f - gate) * h[t + 256] + gate * m[t + 256];
  h[t] = a0;
  h[t + 256] = a1;
  hnl[(size_t)row * 512 + t] = a0;
  hnl[(size_t)row * 512 + t + 256] = a1;
}

// ---------------------------------------------------------------------------
// Final vocab projection: y = z(64,512) @ Wh^T(512,50257) + bh.
// One wave per N-strip of 16, all 4 M-tiles stacked (B fragment reused 4x).
// Ntiles = ceil(50257/16) = 3142; grid = 393 blocks x 8 waves.
// ---------------------------------------------------------------------------
__global__ void k_final(const float* __restrict__ z,
                        const float* __restrict__ Wh,
                        const float* __restrict__ bh,
                        float* __restrict__ y) {
  int wv = threadIdx.x >> 5, lane = threadIdx.x & 31;
  int nt = blockIdx.x * 8 + wv;
  if (nt >= 3142) return;  // wave-uniform: EXEC stays all-ones for live waves
  int sub = lane & 15, half = lane >> 4;
  int n = nt * 16 + sub;
  int ncl = n < 50257 ? n : 50256;  // clamp OOB rows for the load
  v8f zero = {};
  v8f acc[4] = {zero, zero, zero, zero};
  strip4_single(z + sub * 512 + 2 * half,
                Wh + (size_t)ncl * 512 + 2 * half, acc);
  if (n < 50257) {
    float bb = bh[n];
#pragma unroll
    for (int mt = 0; mt < 4; ++mt)
#pragma unroll
      for (int r = 0; r < 8; ++r)
        y[(size_t)(mt * 16 + r + 8 * half) * 50257 + n] = acc[mt][r] + bb;
  }
}

// ---------------------------------------------------------------------------
// Host orchestration
// ---------------------------------------------------------------------------
extern "C" void kernel_launch(void* const* d_in, const int* in_sizes, int n_in,
                              void* d_out, int out_size, void* d_ws,
                              size_t ws_size, hipStream_t stream) {
  const int* tokens   = (const int*)d_in[0];
  const float* h0     = (const float*)d_in[1];
  const float* c0     = (const float*)d_in[2];
  const float* emb    = (const float*)d_in[3];
  const float* Wih0   = (const float*)d_in[4];
  const float* Whh0   = (const float*)d_in[5];
  const float* bih0   = (const float*)d_in[6];
  const float* bhh0   = (const float*)d_in[7];
  // d_in[8] = Wih0r : unused in the forward pass
  const float* Whh0r  = (const float*)d_in[9];
  const float* bih0r  = (const float*)d_in[10];
  const float* bhh0r  = (const float*)d_in[11];
  const float* Wih    = (const float*)d_in[12];
  const float* Whh    = (const float*)d_in[13];
  const float* bih    = (const float*)d_in[14];
  const float* bhh    = (const float*)d_in[15];
  const float* Wq     = (const float*)d_in[16];
  const float* Wk     = (const float*)d_in[17];
  const float* Wv     = (const float*)d_in[18];
  const float* Wo     = (const float*)d_in[19];
  const float* bo     = (const float*)d_in[20];
  const float* logb   = (const float*)d_in[21];
  const float* ln_g   = (const float*)d_in[22];
  const float* ln_b   = (const float*)d_in[23];
  const float* Wg     = (const float*)d_in[24];
  const float* bg     = (const float*)d_in[25];
  const float* Wh     = (const float*)d_in[26];
  const float* bh     = (const float*)d_in[27];

  const size_t CBH = 8 * 64 * 512;       // 262144
  float* out = (float*)d_out;
  float* y   = out;                       // (B, O)
  float* hn  = out + (size_t)64 * 50257;  // (L,C,B,H)
  float* cn  = hn + 4 * CBH;              // (L,C,B,H)

  float* ws    = (float*)d_ws;
  float* x     = ws;                 // 64*512
  float* hl    = x + 64 * 512;       // CBH
  float* gates = hl + CBH;           // 8*64*2048
  float* q     = gates + 8 * 64 * 2048;
  float* kb    = q + CBH;
  float* vb    = kb + CBH;
  float* ao    = vb + CBH;
  float* op    = ao + CBH;
  float* msg   = op + CBH;

  // Embedding
  k_embed<<<128, 256, 0, stream>>>(tokens, emb, x);

  // Layer 0 cells
  k_gates_l0<<<dim3(16, 8), 256, 0, stream>>>(x, h0, Wih0, Whh0, bih0, bhh0,
                                              Whh0r, bih0r, bhh0r, gates);
  k_lstm<<<1024, 256, 0, stream>>>(gates, c0, hl, cn);

  for (int l = 0; l < 4; ++l) {
    if (l > 0) {
      k_gates_dual<<<dim3(16, 8), 256, 0, stream>>>(
          hl, h0 + (size_t)l * CBH,
          Wih + (size_t)(l - 1) * 8 * 2048 * 512,
          Whh + (size_t)(l - 1) * 8 * 2048 * 512,
          bih + (size_t)(l - 1) * 8 * 2048,
          bhh + (size_t)(l - 1) * 8 * 2048, gates);
      k_lstm<<<1024, 256, 0, stream>>>(gates, c0 + (size_t)l * CBH, hl,
                                       cn + (size_t)l * CBH);
    }
    // Hopfield projections (512x512x512 each), q/k/v in one launch
    k_qkv<<<dim3(32, 3), 256, 0, stream>>>(hl, Wq + (size_t)l * 512 * 512,
                                           Wk + (size_t)l * 512 * 512,
                                           Wv + (size_t)l * 512 * 512, q, kb,
                                           vb);
    k_attn<<<512, 64, 0, stream>>>(q, kb, vb, logb + l * 8, ao);
    k_proj512<<<32, 256, 0, stream>>>(ao, Wo + (size_t)l * 512 * 512,
                                      bo + l * 512, op);
    k_ln<<<512, 256, 0, stream>>>(op, ln_g + l * 512, ln_b + l * 512, msg);
    k_mix<<<512, 256, 0, stream>>>(hl, msg, Wg + l * 1024, bg + l,
                                   hn + (size_t)l * CBH);
  }

  // y = h_n[3][0] @ Wh^T + bh   (z = first 64 rows of hl, i.e. column 0)
  k_final<<<393, 256, 0, stream>>>(hl, Wh, bh, y);
}